// HMQAct_59983513256148
// MI455X (gfx1250) — compile-verified
//
#include <hip/hip_runtime.h>

// Block floating-point quantization for MI455X (gfx1250, wave32).
//
// x: (4, 4096, 4096) f32; 16 contiguous elements share floor(log2(max|x|));
// 8-bit signed mantissa (qmin=-128, qmax=127), round-half-to-even, rescale.
//
// Roofline: 256 MB in + 256 MB out = 512 MB >> 192 MB L2 -> pure streaming.
// Non-temporal 128-bit loads/stores; one lane owns a float4; a 16-elem block
// is 4 consecutive lanes reduced with two single-instruction ds_swizzle ops
// (SWAPX1 / SWAPX2 — xor masks 1 and 2 never leave the wave, so no lane
// clamping needed, unlike the generic __shfl_xor path).

typedef __attribute__((ext_vector_type(4))) float v4f;

#define MBITS 8
#define QMAXF 127.0f
#define QMINF -128.0f

// ds_swizzle group-of-32 encoding: offset[15]=0, xor=offset[14:10],
// or=offset[9:5], and=offset[4:0].  SWAPX1: xor=1 -> 0x041F; SWAPX2 -> 0x081F.
__device__ __forceinline__ float swizzle_xor1(float v) {
    int r = __builtin_amdgcn_ds_swizzle(__builtin_bit_cast(int, v), 0x041F);
    return __builtin_bit_cast(float, r);
}
__device__ __forceinline__ float swizzle_xor2(float v) {
    int r = __builtin_amdgcn_ds_swizzle(__builtin_bit_cast(int, v), 0x081F);
    return __builtin_bit_cast(float, r);
}

__global__ __launch_bounds__(256) void bfp_quant_kernel(
    const v4f* __restrict__ x, v4f* __restrict__ out, int nvec)
{
    const int tid    = blockIdx.x * blockDim.x + threadIdx.x;
    const int stride = gridDim.x * blockDim.x;

    for (int i = tid; i < nvec; i += stride) {
        // Stage lines ~4 strides ahead into L2 (global_prefetch_b8) to keep
        // the HBM pipe deep; condition is wave-uniform given the launch shape.
        if (i + 4 * stride < nvec) {
            __builtin_prefetch((const void*)&x[i + 4 * stride], 0, 1);
        }

        // Streaming 128-bit load (TH=NT: read-once data, don't pollute L2).
        v4f v = __builtin_nontemporal_load(&x[i]);

        // Per-lane max|.| over the 4 owned elements.
        float m = fmaxf(fmaxf(__builtin_fabsf(v.x), __builtin_fabsf(v.y)),
                        fmaxf(__builtin_fabsf(v.z), __builtin_fabsf(v.w)));

        // Block max across the xor-group of 4 lanes (16 elements total):
        // two ds_swizzle_b32 on the LDS crossbar, zero memory traffic.
        m = fmaxf(m, swizzle_xor1(m));
        m = fmaxf(m, swizzle_xor2(m));

        // shared_exp = floor(log2(safe_max)). frexpf: safe = f * 2^e with
        // f in [0.5,1), so floor(log2(safe)) = e - 1 (denormal-safe).
        const float safe = (m > 0.0f) ? m : 1.0f;
        int e;
        (void)frexpf(safe, &e);
        const int sexp = e - 1;

        // scale = 2^(sexp-(MBITS-1)); scale and rscale are exact powers of
        // two, so multiply-by-reciprocal is bit-identical to the divide.
        const float scale  = ldexpf(1.0f, sexp - (MBITS - 1));
        const float rscale = ldexpf(1.0f, (MBITS - 1) - sexp);

        v4f q;
        #pragma unroll
        for (int k = 0; k < 4; ++k) {
            float r = rintf(v[k] * rscale);        // round half-to-even (RTNE)
            r = fminf(fmaxf(r, QMINF), QMAXF);     // clip to signed 8-bit range
            q[k] = r * scale;
        }
        // max_abs == 0 => all 16 elements are 0 => q is exactly 0 already,
        // matching the reference's where(max_abs > 0, q, 0).

        // Streaming 128-bit store (TH=NT).
        __builtin_nontemporal_store(q, &out[i]);
    }
}

extern "C" void kernel_launch(void* const* d_in, const int* in_sizes, int n_in,
                              void* d_out, int out_size, void* d_ws, size_t ws_size,
                              hipStream_t stream)
{
    const float* x  = (const float*)d_in[0];
    float*      out = (float*)d_out;

    const int n    = in_sizes[0];   // 4*4096*4096 = 67,108,864 (multiple of 16)
    const int nvec = n / 4;         // float4 granules

    const int threads = 256;        // 8 wave32 waves per block
    int blocks = (nvec + threads - 1) / threads;
    if (blocks > 16384) blocks = 16384;  // grid-stride, ~4 iterations/thread

    bfp_quant_kernel<<<blocks, threads, 0, stream>>>(
        (const v4f*)x, (v4f*)out, nvec);
}